// ModernGNNBlock_81793357185797
// MI455X (gfx1250) — compile-verified
//
#include <hip/hip_runtime.h>

typedef __attribute__((ext_vector_type(2))) float v2f;
typedef __attribute__((ext_vector_type(8))) float v8f;

#define C_DIM 128
#define LN_EPS 1e-5f

// ---------------------------------------------------------------------------
// Kernel 1: LayerNorm + ReLU, one wave32 per row of x[N, 128].
// Each lane owns 4 contiguous channels (float4); mean/var via wave shuffles.
// ---------------------------------------------------------------------------
__global__ void __launch_bounds__(256)
ln_relu_kernel(const float* __restrict__ x, const float* __restrict__ gamma,
               const float* __restrict__ beta, float* __restrict__ h, int N) {
  int row  = (int)((blockIdx.x * (unsigned)blockDim.x + threadIdx.x) >> 5);
  int lane = threadIdx.x & 31;
  if (row >= N) return;

  const float4 v = ((const float4*)(x + (size_t)row * C_DIM))[lane];
  float s  = v.x + v.y + v.z + v.w;
  float ss = v.x * v.x + v.y * v.y + v.z * v.z + v.w * v.w;
#pragma unroll
  for (int m = 16; m >= 1; m >>= 1) {
    s  += __shfl_xor(s, m, 32);
    ss += __shfl_xor(ss, m, 32);
  }
  const float mu  = s * (1.0f / C_DIM);
  const float var = ss * (1.0f / C_DIM) - mu * mu;
  const float rs  = rsqrtf(var + LN_EPS);

  const int c = lane * 4;
  const float4 g = *(const float4*)(gamma + c);
  const float4 b = *(const float4*)(beta + c);
  float4 o;
  o.x = fmaxf(0.0f, (v.x - mu) * rs * g.x + b.x);
  o.y = fmaxf(0.0f, (v.y - mu) * rs * g.y + b.y);
  o.z = fmaxf(0.0f, (v.z - mu) * rs * g.z + b.z);
  o.w = fmaxf(0.0f, (v.w - mu) * rs * g.w + b.w);
  ((float4*)(h + (size_t)row * C_DIM))[lane] = o;
}

// ---------------------------------------------------------------------------
// Kernel 2: zero the accumulator + count region (required every launch:
// the graph replays and the harness does not re-poison workspace).
// ---------------------------------------------------------------------------
__global__ void zero_kernel(float* __restrict__ p, size_t n) {
  size_t i = (size_t)blockIdx.x * blockDim.x + threadIdx.x;
  if (i < n) p[i] = 0.0f;
}

// ---------------------------------------------------------------------------
// Kernel 3: edge scatter-add. One wave32 per edge; each lane moves 4 channels
// of h[src] into agg[dst] via global_atomic_add_f32. Lane 0 counts the edge.
// 512 B of payload per edge -- the minimum possible data movement.
// ---------------------------------------------------------------------------
__global__ void __launch_bounds__(256)
scatter_kernel(const long long* __restrict__ ei, const float* __restrict__ h,
               float* __restrict__ agg, float* __restrict__ cnt, long long E) {
  long long e = (((long long)blockIdx.x * blockDim.x) + threadIdx.x) >> 5;
  int lane = threadIdx.x & 31;
  if (e >= E) return;
  const long long src = ei[e];       // scalar (uniform within wave)
  const long long dst = ei[E + e];

  const float4 v = ((const float4*)(h + src * C_DIM))[lane];
  float* a = agg + dst * C_DIM + lane * 4;
  atomicAdd(a + 0, v.x);
  atomicAdd(a + 1, v.y);
  atomicAdd(a + 2, v.z);
  atomicAdd(a + 3, v.w);
  if (lane == 0) atomicAdd(cnt + dst, 1.0f);
}

// ---------------------------------------------------------------------------
// Kernel 4: mean = agg / max(cnt, 1), in place (agg==0 when cnt==0, matching
// the reference's where()).
// ---------------------------------------------------------------------------
__global__ void mean_kernel(float* __restrict__ agg, const float* __restrict__ cnt,
                            int total) {
  int i = blockIdx.x * blockDim.x + threadIdx.x;
  if (i >= total) return;
  agg[i] = agg[i] / fmaxf(cnt[i >> 7], 1.0f);
}

// ---------------------------------------------------------------------------
// Kernel 5: fused double GEMM + bias + residual using V_WMMA_F32_16X16X4_F32.
//   out = mean @ Wl^T + h @ Wr^T + bl + x
// Block = 256 threads = 8 waves; one 16-row tile per block, wave w owns
// output columns [16w, 16w+16). K loop steps by 4; both GEMMs share one
// fp32 accumulator (exact precision, and the op is nowhere near any roofline).
//
// Fragment layouts per CDNA5 ISA 7.12.2:
//   A 16x4 : lane<16 -> row=lane,    v0=K+0, v1=K+1
//            lane>=16-> row=lane-16, v0=K+2, v1=K+3        (float2 load)
//   B 4x16 : lane<16 -> col=lane,    v0=K+0, v1=K+1  (W row-major [n,k] IS
//            lane>=16-> col=lane-16, v0=K+2, v1=K+3   the K x N transpose)
//   C/D    : vgpr r -> row = r + (lane<16 ? 0 : 8), col = lane & 15
// ---------------------------------------------------------------------------
__global__ void __launch_bounds__(256)
gemm_wmma_kernel(const float* __restrict__ mean, const float* __restrict__ h,
                 const float* __restrict__ Wl, const float* __restrict__ Wr,
                 const float* __restrict__ bl, const float* __restrict__ x,
                 float* __restrict__ out) {
  const int i0   = blockIdx.x * 16;       // row tile
  const int wave = threadIdx.x >> 5;      // 0..7 -> column tile
  const int lane = threadIdx.x & 31;
  const int n0   = wave * 16;
  const int m    = lane & 15;
  const int kk   = (lane >> 4) << 1;      // 0 or 2

  const float* meanRow = mean + (size_t)(i0 + m) * C_DIM + kk;
  const float* hRow    = h    + (size_t)(i0 + m) * C_DIM + kk;
  const float* wlRow   = Wl   + (size_t)(n0 + m) * C_DIM + kk;
  const float* wrRow   = Wr   + (size_t)(n0 + m) * C_DIM + kk;

  v8f acc = {};
#pragma unroll
  for (int k0 = 0; k0 < C_DIM; k0 += 4) {
    const v2f aM = *(const v2f*)(meanRow + k0);
    const v2f bL = *(const v2f*)(wlRow + k0);
    acc = __builtin_amdgcn_wmma_f32_16x16x4_f32(false, aM, false, bL,
                                                (short)0, acc, false, false);
    const v2f aH = *(const v2f*)(hRow + k0);
    const v2f bR = *(const v2f*)(wrRow + k0);
    acc = __builtin_amdgcn_wmma_f32_16x16x4_f32(false, aH, false, bR,
                                                (short)0, acc, false, false);
  }

  const int col   = n0 + m;
  const float bias = bl[col];
  const int rbase = i0 + ((lane >> 4) << 3);
#pragma unroll
  for (int r = 0; r < 8; ++r) {
    const size_t idx = (size_t)(rbase + r) * C_DIM + col;
    out[idx] = acc[r] + bias + x[idx];
  }
}

// Scalar tail for rows beyond the last full 16-row tile (N=100000 divides
// evenly, but dims are derived from in_sizes so keep it robust).
__global__ void tail_kernel(const float* __restrict__ mean, const float* __restrict__ h,
                            const float* __restrict__ Wl, const float* __restrict__ Wr,
                            const float* __restrict__ bl, const float* __restrict__ x,
                            float* __restrict__ out, int rowStart, int N) {
  int idx = blockIdx.x * blockDim.x + threadIdx.x;
  int row = rowStart + (idx >> 7);
  int col = idx & 127;
  if (row >= N) return;
  const float* mr = mean + (size_t)row * C_DIM;
  const float* hr = h + (size_t)row * C_DIM;
  const float* wl = Wl + (size_t)col * C_DIM;
  const float* wr = Wr + (size_t)col * C_DIM;
  float acc = bl[col];
  for (int k = 0; k < C_DIM; ++k) acc += mr[k] * wl[k] + hr[k] * wr[k];
  out[(size_t)row * C_DIM + col] = acc + x[(size_t)row * C_DIM + col];
}

extern "C" void kernel_launch(void* const* d_in, const int* in_sizes, int n_in,
                              void* d_out, int out_size, void* d_ws, size_t ws_size,
                              hipStream_t stream) {
  const float*     x     = (const float*)d_in[0];
  const long long* ei    = (const long long*)d_in[1];   // int64 edge_index
  const float*     gamma = (const float*)d_in[2];
  const float*     beta  = (const float*)d_in[3];
  const float*     Wl    = (const float*)d_in[4];
  const float*     bl    = (const float*)d_in[5];
  const float*     Wr    = (const float*)d_in[6];
  float*           out   = (float*)d_out;

  const int       N = in_sizes[0] / C_DIM;
  const long long E = (long long)in_sizes[1] / 2;

  float* h   = (float*)d_ws;                 // [N, 128]
  float* agg = h + (size_t)N * C_DIM;        // [N, 128], becomes `mean` in place
  float* cnt = agg + (size_t)N * C_DIM;      // [N]

  // 1) LayerNorm + ReLU (8 rows per 256-thread block)
  ln_relu_kernel<<<(N + 7) / 8, 256, 0, stream>>>(x, gamma, beta, h, N);

  // 2) Zero agg + cnt
  const size_t nz = (size_t)N * C_DIM + (size_t)N;
  zero_kernel<<<(unsigned)((nz + 255) / 256), 256, 0, stream>>>(agg, nz);

  // 3) Edge scatter-add (8 edges per block)
  scatter_kernel<<<(unsigned)((E + 7) / 8), 256, 0, stream>>>(ei, h, agg, cnt, E);

  // 4) agg -> mean in place
  const int total = N * C_DIM;
  mean_kernel<<<(total + 255) / 256, 256, 0, stream>>>(agg, cnt, total);

  // 5) Fused WMMA double-GEMM + bias + residual
  const int Mt = N / 16;
  if (Mt > 0)
    gemm_wmma_kernel<<<Mt, 256, 0, stream>>>(agg, h, Wl, Wr, bl, x, out);
  const int rem = N - Mt * 16;
  if (rem > 0)
    tail_kernel<<<(rem * C_DIM + 255) / 256, 256, 0, stream>>>(
        agg, h, Wl, Wr, bl, x, out, Mt * 16, N);
}